// SequentialClusteringModule_395136991788
// MI455X (gfx1250) — compile-verified
//
#include <hip/hip_runtime.h>
#include <math.h>

// Problem constants (from reference)
#define EPS_      1e-5f
#define B_        128
#define T_        300
#define DV_       1024
#define DA_       128
#define KSEL_     32            // CLUSTER_COUNT
#define NSEL_     (KSEL_ + 1)   // 33 output rows per batch
#define NPAIR_    (T_ - 1)      // 299 adjacent pairs
#define PPB_      15            // pairs covered per 16-row tile
#define NBLK_     20            // ceil(299/15)

typedef __attribute__((ext_vector_type(2))) float v2f;
typedef __attribute__((ext_vector_type(8))) float v8f;

// ---------------------------------------------------------------------------
// Kernel 1: per (batch, 16-row tile) Gram matrix via V_WMMA_F32_16X16X4_F32.
//
// Column->K-slot mapping (order-free since Gram sums over all columns, and
// A/B share lanes+layout): hi=0 lanes supply K={0,1} from columns 0..511,
// hi=1 lanes supply K={2,3} from columns 512..1023. Each lane's data is then
// a sequential stream -> b128 loads, one float4 = two K-chunks.
//
// Tiles: t0 = min(15*blk, 284) so all 16 rows are in-bounds (no padding);
// the one overlapping pair is recomputed bit-identically. Diagonal of G ->
// squared norms, super-diagonal -> adjacent dots.
// ---------------------------------------------------------------------------
__global__ __launch_bounds__(32)
void sim_wmma_kernel(const float* __restrict__ video, float* __restrict__ sim)
{
    const int blk  = blockIdx.x % NBLK_;
    const int b    = blockIdx.x / NBLK_;
    int t0 = blk * PPB_;
    if (t0 > T_ - 16) t0 = T_ - 16;   // clamp: rows t0..t0+15 always valid

    const int lane = threadIdx.x;     // 0..31 (wave32)
    const int m    = lane & 15;       // tile row
    const int hi   = lane >> 4;       // 0: columns [0,512), 1: columns [512,1024)
    const int row  = t0 + m;

    const float* p = video + ((size_t)b * T_ + row) * DV_ + hi * (DV_ / 2);

    v8f c0 = {};
    v8f c1 = {};
    // 512 floats per lane, 16 per iteration (4 x b128) -> 32 iters, 8 wmma each
    for (int k = 0; k < DV_ / 2; k += 16) {
        float4 q0 = *(const float4*)(p + k);
        float4 q1 = *(const float4*)(p + k + 4);
        float4 q2 = *(const float4*)(p + k + 8);
        float4 q3 = *(const float4*)(p + k + 12);
        v2f a0 = {q0.x, q0.y}, b0 = {q0.z, q0.w};
        v2f a1 = {q1.x, q1.y}, b1 = {q1.z, q1.w};
        v2f a2 = {q2.x, q2.y}, b2 = {q2.z, q2.w};
        v2f a3 = {q3.x, q3.y}, b3 = {q3.z, q3.w};
        c0 = __builtin_amdgcn_wmma_f32_16x16x4_f32(false, a0, false, a0, (short)0, c0, false, false);
        c1 = __builtin_amdgcn_wmma_f32_16x16x4_f32(false, b0, false, b0, (short)0, c1, false, false);
        c0 = __builtin_amdgcn_wmma_f32_16x16x4_f32(false, a1, false, a1, (short)0, c0, false, false);
        c1 = __builtin_amdgcn_wmma_f32_16x16x4_f32(false, b1, false, b1, (short)0, c1, false, false);
        c0 = __builtin_amdgcn_wmma_f32_16x16x4_f32(false, a2, false, a2, (short)0, c0, false, false);
        c1 = __builtin_amdgcn_wmma_f32_16x16x4_f32(false, b2, false, b2, (short)0, c1, false, false);
        c0 = __builtin_amdgcn_wmma_f32_16x16x4_f32(false, a3, false, a3, (short)0, c0, false, false);
        c1 = __builtin_amdgcn_wmma_f32_16x16x4_f32(false, b3, false, b3, (short)0, c1, false, false);
    }
    v8f g = c0 + c1;

    // Spill C/D tile to LDS: VGPR r holds G[M][N] with M = hi*8 + r, N = m
    __shared__ float G[16 * 16];
    const int mbase = hi * 8;
#pragma unroll
    for (int r = 0; r < 8; ++r)
        G[(mbase + r) * 16 + m] = g[r];
    __syncthreads();

    // lanes 0..14 emit sim for pair (t, t+1), t = t0 + lane
    const int t = t0 + lane;
    if (lane < PPB_ && t < NPAIR_) {
        const float n0 = sqrtf(G[lane * 16 + lane]) + EPS_;
        const float n1 = sqrtf(G[(lane + 1) * 16 + (lane + 1)]) + EPS_;
        const float d  = G[lane * 16 + (lane + 1)];
        sim[(size_t)b * NPAIR_ + t] = fabsf(d) / (n0 * n1);
    }
}

// ---------------------------------------------------------------------------
// Kernel 2: exact top-k (k smallest sims, ascending, ties -> lower index)
// via rank selection. Ranks form a total order -> conflict-free scatter.
// Emits idx[b][0]=0, idx[b][1+rank]=t+1.
// ---------------------------------------------------------------------------
__global__ __launch_bounds__(128)
void topk_kernel(const float* __restrict__ sim, int* __restrict__ idx)
{
    const int b = blockIdx.x;
    __shared__ float s[NPAIR_];
    for (int t = threadIdx.x; t < NPAIR_; t += blockDim.x)
        s[t] = sim[(size_t)b * NPAIR_ + t];
    __syncthreads();

    if (threadIdx.x == 0)
        idx[b * NSEL_] = 0;

    for (int t = threadIdx.x; t < NPAIR_; t += blockDim.x) {
        const float v = s[t];
        int rank = 0;
        for (int u = 0; u < NPAIR_; ++u) {
            const float w = s[u];
            rank += (w < v) || ((w == v) && (u < t));
        }
        if (rank < KSEL_)
            idx[b * NSEL_ + 1 + rank] = t + 1;
    }
}

// ---------------------------------------------------------------------------
// Kernel 3: gather 33 selected rows per batch for video (1024 f32) and audio
// (128 f32) into the concatenated output, fully coalesced b128 copies.
// ---------------------------------------------------------------------------
__global__ __launch_bounds__(256)
void gather_kernel(const float* __restrict__ video,
                   const float* __restrict__ audio,
                   const int* __restrict__ idx,
                   float* __restrict__ out)
{
    const int j = blockIdx.x % NSEL_;
    const int b = blockIdx.x / NSEL_;
    const int t = idx[b * NSEL_ + j];
    const int tid = threadIdx.x;

    const float4* sv = (const float4*)(video + ((size_t)b * T_ + t) * DV_);
    const float4* sa = (const float4*)(audio + ((size_t)b * T_ + t) * DA_);
    float4* dv = (float4*)(out + ((size_t)b * NSEL_ + j) * DV_);
    float4* da = (float4*)(out + (size_t)B_ * NSEL_ * DV_
                               + ((size_t)b * NSEL_ + j) * DA_);

    dv[tid] = sv[tid];               // 256 threads * 16B = 4KB video row
    if (tid < DA_ / 4)
        da[tid] = sa[tid];           // 32 threads * 16B = 512B audio row
}

// ---------------------------------------------------------------------------
extern "C" void kernel_launch(void* const* d_in, const int* in_sizes, int n_in,
                              void* d_out, int out_size, void* d_ws, size_t ws_size,
                              hipStream_t stream)
{
    (void)in_sizes; (void)n_in; (void)out_size; (void)ws_size;
    const float* video = (const float*)d_in[0];   // [128,300,1024] f32
    const float* audio = (const float*)d_in[1];   // [128,300,128]  f32
    float* out = (float*)d_out;                   // [128,33,1024] ++ [128,33,128]

    // workspace layout: sim buffer then (256B-aligned) index buffer
    float* sim = (float*)d_ws;
    size_t sim_bytes = ((size_t)B_ * NPAIR_ * sizeof(float) + 255) & ~(size_t)255;
    int* idx = (int*)((char*)d_ws + sim_bytes);

    sim_wmma_kernel<<<B_ * NBLK_, 32, 0, stream>>>(video, sim);
    topk_kernel<<<B_, 128, 0, stream>>>(sim, idx);
    gather_kernel<<<B_ * NSEL_, 256, 0, stream>>>(video, audio, idx, out);
}